// TopKGate_13288628813931
// MI455X (gfx1250) — compile-verified
//
#include <hip/hip_runtime.h>
#include <hip/hip_bf16.h>

typedef __attribute__((ext_vector_type(2))) float v2f;
typedef __attribute__((ext_vector_type(8))) float v8f;

#define MODEL_DIM   4096
#define PROMPT_DIM  64
#define GATE_DIM    (MODEL_DIM + PROMPT_DIM)     /* 4160 */
#define NUM_EXPERTS 8
#define NPAD        16                           /* experts padded to WMMA N */
#define TILE_M      16
#define KSPLIT      4                            /* waves cooperating on one tile */
#define TILES_PER_BLOCK 2
#define XCHUNK      (MODEL_DIM / KSPLIT)         /* 1024 x-cols per wave (const) */
#define PCHUNK      (PROMPT_DIM / KSPLIT)        /* 16 prompt-cols per wave (const) */
#define LDS_STRIDE  17                           /* 16 + 1 pad: no bank conflicts */
#define EPS_F       1.1920928955078125e-7f       /* np.float32 eps */
#define WT_ELEMS    (NPAD * GATE_DIM)            /* 66560 floats = 266240 B */

// ---------------------------------------------------------------------------
// Prep: Wt[e][k] = (e < 8) ? W[k][e] : 0   (column-major W, zero-padded to 16)
// Makes the per-lane B fragment a contiguous float2 -> one global_load_b64.
// ---------------------------------------------------------------------------
__global__ __launch_bounds__(256)
void build_wt(const float* __restrict__ W, float* __restrict__ Wt)
{
    int idx = blockIdx.x * blockDim.x + threadIdx.x;   // e*4160 + k
    if (idx >= WT_ELEMS) return;
    int e = idx / GATE_DIM;
    int k = idx - e * GATE_DIM;
    Wt[idx] = (e < NUM_EXPERTS) ? W[(size_t)k * NUM_EXPERTS + e] : 0.f;
}

// ---------------------------------------------------------------------------
// Shared post-processing: softmax over 8 experts, top-2, renorm, outputs.
// ---------------------------------------------------------------------------
__device__ __forceinline__
void postprocess(const float* tl, const float* __restrict__ bias,
                 int t, float* __restrict__ out_masks, float* __restrict__ out_gates,
                 int tok)
{
    float lg[NUM_EXPERTS];
    #pragma unroll
    for (int e = 0; e < NUM_EXPERTS; ++e)
        lg[e] = tl[tok * LDS_STRIDE + e] + bias[e];

    // top-1 / top-2, first-occurrence tie-break (matches jax.lax.top_k)
    int i0 = 0; float v0 = lg[0];
    #pragma unroll
    for (int e = 1; e < NUM_EXPERTS; ++e)
        if (lg[e] > v0) { v0 = lg[e]; i0 = e; }
    int i1 = (i0 == 0) ? 1 : 0; float v1 = lg[i1];
    #pragma unroll
    for (int e = 0; e < NUM_EXPERTS; ++e)
        if (e != i0 && lg[e] > v1) { v1 = lg[e]; i1 = e; }

    float sum = 0.f;
    #pragma unroll
    for (int e = 0; e < NUM_EXPERTS; ++e)
        sum += __expf(lg[e] - v0);
    const float inv = 1.f / sum;
    const float s0  = inv;                       // exp(0)/sum
    const float s1  = __expf(v1 - v0) * inv;
    float denom = s0 + s1;
    denom = denom > EPS_F ? denom : EPS_F;
    const float g0 = s0 / denom;
    const float g1 = s1 / denom;

    float mk[2 * NUM_EXPERTS];
    #pragma unroll
    for (int e = 0; e < NUM_EXPERTS; ++e) {
        mk[e]               = (e == i0) ? 1.f : 0.f;
        mk[NUM_EXPERTS + e] = (e == i1) ? 1.f : 0.f;
    }
    float* mo = out_masks + (size_t)t * (2 * NUM_EXPERTS);
    #pragma unroll
    for (int q = 0; q < 4; ++q) {
        float4 v = make_float4(mk[4*q+0], mk[4*q+1], mk[4*q+2], mk[4*q+3]);
        *(float4*)(mo + 4 * q) = v;
    }
    out_gates[(size_t)t * 2 + 0] = g0;
    out_gates[(size_t)t * 2 + 1] = g1;
}

// ---------------------------------------------------------------------------
// Fast path: B fragments from pre-transposed, zero-padded Wt.
// All loops have literal-constant trip counts -> full unroll/pipelining.
// ---------------------------------------------------------------------------
__global__ __launch_bounds__(TILES_PER_BLOCK * KSPLIT * 32)
void topk_gate_wmma_fast(const float* __restrict__ x,
                         const float* __restrict__ prompt,
                         const float* __restrict__ Wt,
                         const float* __restrict__ bias,
                         float* __restrict__ out_masks,
                         float* __restrict__ out_gates)
{
    __shared__ float tileAcc[TILES_PER_BLOCK][TILE_M * LDS_STRIDE];

    const int tid   = threadIdx.x;
    const int lane  = tid & 31;
    const int wave  = tid >> 5;
    const int tileL = wave >> 2;
    const int chunk = wave & (KSPLIT - 1);
    const int row0  = (blockIdx.x * TILES_PER_BLOCK + tileL) * TILE_M;

    for (int i = tid; i < TILES_PER_BLOCK * TILE_M * LDS_STRIDE; i += blockDim.x)
        (&tileAcc[0][0])[i] = 0.f;
    __syncthreads();

    // WMMA f32 16x16x4 lane mapping:
    // A: lanes 0-15 row M=lane K0/K1, lanes 16-31 K2/K3
    // B: lanes 0-15 col N=lane K0/K1, lanes 16-31 K2/K3
    const int m    = lane & 15;
    const int kOfs = (lane >> 4) * 2;

    // per-wave constant-stride base pointers (loop offsets are literals)
    const float* __restrict__ aPtr  = x + (size_t)(row0 + m) * MODEL_DIM
                                        + chunk * XCHUNK + kOfs;
    const float* __restrict__ wPtr  = Wt + (size_t)m * GATE_DIM
                                        + chunk * XCHUNK + kOfs;
    const float* __restrict__ pPtr  = prompt + (size_t)(row0 + m) * PROMPT_DIM
                                        + chunk * PCHUNK + kOfs;
    const float* __restrict__ w2Ptr = Wt + (size_t)m * GATE_DIM + MODEL_DIM
                                        + chunk * PCHUNK + kOfs;

    v8f acc = {};

    // ---- x part: constant 256 steps, unroll x8 (no remainder) ----
    #pragma unroll 8
    for (int s = 0; s < XCHUNK; s += 4) {
        v2f a = *(const v2f*)(aPtr + s);
        v2f b = *(const v2f*)(wPtr + s);
        acc = __builtin_amdgcn_wmma_f32_16x16x4_f32(
                  false, a, false, b, (short)0, acc, false, false);
    }
    // ---- prompt part: constant 4 steps, fully unrolled ----
    #pragma unroll
    for (int s = 0; s < PCHUNK; s += 4) {
        v2f a = *(const v2f*)(pPtr + s);
        v2f b = *(const v2f*)(w2Ptr + s);
        acc = __builtin_amdgcn_wmma_f32_16x16x4_f32(
                  false, a, false, b, (short)0, acc, false, false);
    }

    {   // combine K-split partials (C: lanes 0-15 rows 0..7, lanes 16-31 rows 8..15)
        float* tl = tileAcc[tileL];
        const int rBase = (lane >> 4) * 8;
        #pragma unroll
        for (int r = 0; r < 8; ++r)
            atomicAdd(&tl[(rBase + r) * LDS_STRIDE + m], acc[r]);
    }
    __syncthreads();

    if (tid < TILES_PER_BLOCK * TILE_M) {
        const int tSel = tid >> 4;
        const int tok  = tid & 15;
        const int t = (blockIdx.x * TILES_PER_BLOCK + tSel) * TILE_M + tok;
        postprocess(tileAcc[tSel], bias, t, out_masks, out_gates, tok);
    }
}

// ---------------------------------------------------------------------------
// Fallback (no workspace): unconditional clamped loads, zeroed by 0/1 scale.
// Same constant-trip structure.
// ---------------------------------------------------------------------------
__global__ __launch_bounds__(TILES_PER_BLOCK * KSPLIT * 32)
void topk_gate_wmma_fb(const float* __restrict__ x,
                       const float* __restrict__ prompt,
                       const float* __restrict__ W,
                       const float* __restrict__ bias,
                       float* __restrict__ out_masks,
                       float* __restrict__ out_gates)
{
    __shared__ float tileAcc[TILES_PER_BLOCK][TILE_M * LDS_STRIDE];

    const int tid   = threadIdx.x;
    const int lane  = tid & 31;
    const int wave  = tid >> 5;
    const int tileL = wave >> 2;
    const int chunk = wave & (KSPLIT - 1);
    const int row0  = (blockIdx.x * TILES_PER_BLOCK + tileL) * TILE_M;

    for (int i = tid; i < TILES_PER_BLOCK * TILE_M * LDS_STRIDE; i += blockDim.x)
        (&tileAcc[0][0])[i] = 0.f;
    __syncthreads();

    const int   m      = lane & 15;
    const int   kOfs   = (lane >> 4) * 2;
    const float wScale = (m < NUM_EXPERTS) ? 1.f : 0.f;
    const int   wrMax  = GATE_DIM * NUM_EXPERTS - 1;

    const float* __restrict__ aPtr = x + (size_t)(row0 + m) * MODEL_DIM
                                       + chunk * XCHUNK + kOfs;
    const float* __restrict__ pPtr = prompt + (size_t)(row0 + m) * PROMPT_DIM
                                       + chunk * PCHUNK + kOfs;
    const int kx0 = chunk * XCHUNK + kOfs;               // x-part W row base
    const int kp0 = MODEL_DIM + chunk * PCHUNK + kOfs;   // prompt-part W row base

    v8f acc = {};

    #pragma unroll 8
    for (int s = 0; s < XCHUNK; s += 4) {
        v2f a = *(const v2f*)(aPtr + s);
        int wr0 = (kx0 + s) * NUM_EXPERTS + m;
        int wr1 = wr0 + NUM_EXPERTS;
        wr0 = wr0 < wrMax ? wr0 : wrMax;
        wr1 = wr1 < wrMax ? wr1 : wrMax;
        v2f b;
        b.x = W[wr0] * wScale;
        b.y = W[wr1] * wScale;
        acc = __builtin_amdgcn_wmma_f32_16x16x4_f32(
                  false, a, false, b, (short)0, acc, false, false);
    }
    #pragma unroll
    for (int s = 0; s < PCHUNK; s += 4) {
        v2f a = *(const v2f*)(pPtr + s);
        int wr0 = (kp0 + s) * NUM_EXPERTS + m;
        int wr1 = wr0 + NUM_EXPERTS;
        wr0 = wr0 < wrMax ? wr0 : wrMax;
        wr1 = wr1 < wrMax ? wr1 : wrMax;
        v2f b;
        b.x = W[wr0] * wScale;
        b.y = W[wr1] * wScale;
        acc = __builtin_amdgcn_wmma_f32_16x16x4_f32(
                  false, a, false, b, (short)0, acc, false, false);
    }

    {
        float* tl = tileAcc[tileL];
        const int rBase = (lane >> 4) * 8;
        #pragma unroll
        for (int r = 0; r < 8; ++r)
            atomicAdd(&tl[(rBase + r) * LDS_STRIDE + m], acc[r]);
    }
    __syncthreads();

    if (tid < TILES_PER_BLOCK * TILE_M) {
        const int tSel = tid >> 4;
        const int tok  = tid & 15;
        const int t = (blockIdx.x * TILES_PER_BLOCK + tSel) * TILE_M + tok;
        postprocess(tileAcc[tSel], bias, t, out_masks, out_gates, tok);
    }
}

extern "C" void kernel_launch(void* const* d_in, const int* in_sizes, int n_in,
                              void* d_out, int out_size, void* d_ws, size_t ws_size,
                              hipStream_t stream) {
    (void)n_in; (void)out_size;
    const float* x      = (const float*)d_in[0];
    const float* prompt = (const float*)d_in[1];
    const float* W      = (const float*)d_in[2];
    const float* b      = (const float*)d_in[3];

    const int tokens = in_sizes[0] / MODEL_DIM;                 // 8192
    float* out_masks = (float*)d_out;                           // [T, 2, 8]
    float* out_gates = (float*)d_out + (size_t)tokens * 2 * NUM_EXPERTS; // [T, 2]

    const int blocks  = tokens / (TILE_M * TILES_PER_BLOCK);    // 256
    const int threads = TILES_PER_BLOCK * KSPLIT * 32;          // 256 (8 waves)

    if (ws_size >= (size_t)WT_ELEMS * sizeof(float)) {
        float* Wt = (float*)d_ws;
        build_wt<<<(WT_ELEMS + 255) / 256, 256, 0, stream>>>(W, Wt);
        topk_gate_wmma_fast<<<blocks, threads, 0, stream>>>(x, prompt, Wt, b,
                                                            out_masks, out_gates);
    } else {
        topk_gate_wmma_fb<<<blocks, threads, 0, stream>>>(x, prompt, W, b,
                                                          out_masks, out_gates);
    }
}